// GaussianAdapter_4303557231291
// MI455X (gfx1250) — compile-verified
//
#include <hip/hip_runtime.h>
#include <hip/hip_bf16.h>
#include <stdint.h>

// CDNA5 / gfx1250: wave32, WMMA f32 16x16x4, async global<->LDS staging.

typedef __attribute__((ext_vector_type(2))) float v2f;
typedef __attribute__((ext_vector_type(8))) float v8f;

#define VP_STRIDE 32
#define VP_KINV   0   // 9 floats: inv(intrinsics) row-major
#define VP_MULT   9   // 1 float : 0.1 * sum(K2_inv @ pixel_size)
#define VP_Q      10  // 4 floats: c2w quaternion (w,x,y,z)
#define VP_R      14  // 9 floats: c2w rotation row-major
#define VP_T      23  // 3 floats: translation

// ---- single-instruction hardware approximations (~1 ulp) -------------------
__device__ __forceinline__ float fast_rcp (float x) { return __builtin_amdgcn_rcpf(x); }
__device__ __forceinline__ float fast_rsq (float x) { return __builtin_amdgcn_rsqf(x); }
__device__ __forceinline__ float fast_sqrt(float x) { return __builtin_amdgcn_sqrtf(x); }
__device__ __forceinline__ float fast_exp (float x) {
    return __builtin_amdgcn_exp2f(x * 1.44269504088896340736f);
}

// ---------------------------------------------------------------------------
// Per-view precompute (16 views): analytic 3x3 inverse, rotmat->quat,
// pixel multiplier. One thread per view.
// ---------------------------------------------------------------------------
__global__ void gs_view_precompute(const float* __restrict__ extr,
                                   const float* __restrict__ intr,
                                   const int* __restrict__ Hp,
                                   const int* __restrict__ Wp,
                                   float* __restrict__ vp, int nviews)
{
    int v = threadIdx.x;
    if (v >= nviews) return;
    const float invW = 1.0f / (float)(*Wp);
    const float invH = 1.0f / (float)(*Hp);
    const float* E = extr + v * 16;
    const float* K = intr + v * 9;
    float* o = vp + v * VP_STRIDE;

    // inv(intrinsics) via adjugate
    float a = K[0], b = K[1], c = K[2];
    float d = K[3], e = K[4], f = K[5];
    float g = K[6], h = K[7], i = K[8];
    float cof0 =  (e * i - f * h);
    float cof3 = -(d * i - f * g);
    float cof6 =  (d * h - e * g);
    float det  = a * cof0 + b * cof3 + c * cof6;
    float id   = fast_rcp(det);
    o[VP_KINV + 0] =  cof0 * id;
    o[VP_KINV + 1] = -(b * i - c * h) * id;
    o[VP_KINV + 2] =  (b * f - c * e) * id;
    o[VP_KINV + 3] =  cof3 * id;
    o[VP_KINV + 4] =  (a * i - c * g) * id;
    o[VP_KINV + 5] = -(a * f - c * d) * id;
    o[VP_KINV + 6] =  cof6 * id;
    o[VP_KINV + 7] = -(a * h - b * g) * id;
    o[VP_KINV + 8] =  (a * e - b * d) * id;

    // multiplier = 0.1 * sum_ij(inv(K[:2,:2])[i][j] * pixel_size[j])
    float i2  = fast_rcp(a * e - b * d);
    float k00 = e * i2, k01 = -b * i2, k10 = -d * i2, k11 = a * i2;
    o[VP_MULT] = 0.1f * ((k00 * invW + k01 * invH) + (k10 * invW + k11 * invH));

    // c2w rotation + translation
    float m00 = E[0], m01 = E[1],  m02 = E[2],  t0 = E[3];
    float m10 = E[4], m11 = E[5],  m12 = E[6],  t1 = E[7];
    float m20 = E[8], m21 = E[9],  m22 = E[10], t2 = E[11];
    o[VP_R + 0] = m00; o[VP_R + 1] = m01; o[VP_R + 2] = m02;
    o[VP_R + 3] = m10; o[VP_R + 4] = m11; o[VP_R + 5] = m12;
    o[VP_R + 6] = m20; o[VP_R + 7] = m21; o[VP_R + 8] = m22;
    o[VP_T + 0] = t0;  o[VP_T + 1] = t1;  o[VP_T + 2] = t2;

    // rotmat -> quat (argmax-branch form, matches reference)
    float qa0 = fast_sqrt(fmaxf(1.0f + m00 + m11 + m22, 0.0f));
    float qa1 = fast_sqrt(fmaxf(1.0f + m00 - m11 - m22, 0.0f));
    float qa2 = fast_sqrt(fmaxf(1.0f - m00 + m11 - m22, 0.0f));
    float qa3 = fast_sqrt(fmaxf(1.0f - m00 - m11 + m22, 0.0f));
    int idx = 0; float best = qa0;
    if (qa1 > best) { best = qa1; idx = 1; }
    if (qa2 > best) { best = qa2; idx = 2; }
    if (qa3 > best) { best = qa3; idx = 3; }
    float qw, qx, qy, qz;
    if (idx == 0)      { qw = qa0 * qa0; qx = m21 - m12; qy = m02 - m20; qz = m10 - m01; }
    else if (idx == 1) { qw = m21 - m12; qx = qa1 * qa1; qy = m10 + m01; qz = m02 + m20; }
    else if (idx == 2) { qw = m02 - m20; qx = m10 + m01; qy = qa2 * qa2; qz = m12 + m21; }
    else               { qw = m10 - m01; qx = m20 + m02; qy = m21 + m12; qz = qa3 * qa3; }
    float den = fast_rcp(2.0f * fmaxf(best, 0.1f));
    o[VP_Q + 0] = qw * den; o[VP_Q + 1] = qx * den;
    o[VP_Q + 2] = qy * den; o[VP_Q + 3] = qz * den;
}

// ---------------------------------------------------------------------------
// Main kernel: 1 thread per gaussian, 256 threads (8 waves) per block.
// raw_gaussians (18-float AoS) staged in via global_load_async_to_lds_b128;
// means computed with V_WMMA_F32_16X16X4_F32 (exact in f32) and staged out
// via global_store_async_from_lds_b128.
// ---------------------------------------------------------------------------
__global__ void __launch_bounds__(256)
gs_adapter_kernel(const float* __restrict__ coords,
                  const float* __restrict__ depths,
                  const float* __restrict__ opac,
                  const float* __restrict__ raw,
                  const float* __restrict__ vp,
                  float* __restrict__ out,
                  int perView, int N)
{
    __shared__ float s_raw[256 * 18];       // 18 KB staged input slab
    __shared__ float s_wm[8][32][16];       // 16 KB wmma D redistribution
    __shared__ float s_means[256 * 3];      //  3 KB means store staging

    const int tid  = threadIdx.x;
    const int gid  = blockIdx.x * 256 + tid;
    const int lane = tid & 31;
    const int wv   = tid >> 5;

    // ---- async stage this block's raw_gaussians slab into LDS (coalesced) --
    const float* gsrc = raw + (size_t)blockIdx.x * (256 * 18);
    #pragma unroll 1
    for (int i = tid; i < (256 * 18) / 4; i += 256) {
        uint32_t     loff = (uint32_t)(uintptr_t)(&s_raw[i * 4]);
        const float* ga   = gsrc + (size_t)i * 4;
        asm volatile("global_load_async_to_lds_b128 %0, %1, off"
                     :: "v"(loff), "v"(ga) : "memory");
    }

    // ---- uniform per-view parameters (block-uniform index => scalar loads) -
    const int view = (int)((blockIdx.x * 256u) / (unsigned)perView);
    const float* o = vp + view * VP_STRIDE;
    float Ki0 = o[VP_KINV + 0], Ki1 = o[VP_KINV + 1], Ki2 = o[VP_KINV + 2];
    float Ki3 = o[VP_KINV + 3], Ki4 = o[VP_KINV + 4], Ki5 = o[VP_KINV + 5];
    float Ki6 = o[VP_KINV + 6], Ki7 = o[VP_KINV + 7], Ki8 = o[VP_KINV + 8];
    float mult = o[VP_MULT];
    float qcw = o[VP_Q + 0], qcx = o[VP_Q + 1], qcy = o[VP_Q + 2], qcz = o[VP_Q + 3];
    float R00 = o[VP_R + 0], R01 = o[VP_R + 1], R02 = o[VP_R + 2];
    float R10 = o[VP_R + 3], R11 = o[VP_R + 4], R12 = o[VP_R + 5];
    float R20 = o[VP_R + 6], R21 = o[VP_R + 7], R22 = o[VP_R + 8];
    float t0  = o[VP_T + 0], t1  = o[VP_T + 1], t2  = o[VP_T + 2];

    // ---- per-gaussian camera point ----
    float2 cc    = ((const float2*)coords)[gid];
    float  depth = depths[gid];
    float  dX = Ki0 * cc.x + Ki1 * cc.y + Ki2;
    float  dY = Ki3 * cc.x + Ki4 * cc.y + Ki5;
    float  dZ = Ki6 * cc.x + Ki7 * cc.y + Ki8;
    float  sf = depth * fast_rcp(dZ);
    float  px = dX * sf, py = dY * sf, pz = dZ * sf;

    // ---- means via V_WMMA_F32_16X16X4_F32: D[m][n] = p_m . Ext_row_n -------
    // B (4x16, 2 VGPRs: v0={K0|K2}, v1={K1|K3}); columns n>=3 are zero.
    const int n = lane & 15;
    float e0 = 0.0f, e1 = 0.0f;
    if (lane < 16) {           // K=0 -> Ext[n][0], K=1 -> Ext[n][1]
        e0 = (n == 0) ? R00 : (n == 1) ? R10 : (n == 2) ? R20 : 0.0f;
        e1 = (n == 0) ? R01 : (n == 1) ? R11 : (n == 2) ? R21 : 0.0f;
    } else {                   // K=2 -> Ext[n][2], K=3 -> t[n]
        e0 = (n == 0) ? R02 : (n == 1) ? R12 : (n == 2) ? R22 : 0.0f;
        e1 = (n == 0) ? t0  : (n == 1) ? t1  : (n == 2) ? t2  : 0.0f;
    }
    v2f bop; bop.x = e0; bop.y = e1;
    v8f zero8 = {0.f, 0.f, 0.f, 0.f, 0.f, 0.f, 0.f, 0.f};

    // A (16x4, 2 VGPRs): lanes 0-15 = {x,y}, lanes 16-31 = {z,1} of point (l&15)
    int   s0l = lane & 15;
    float sx = __shfl(px, s0l, 32), sy = __shfl(py, s0l, 32), sz = __shfl(pz, s0l, 32);
    v2f a0; a0.x = (lane < 16) ? sx : sz; a0.y = (lane < 16) ? sy : 1.0f;
    v8f d0 = __builtin_amdgcn_wmma_f32_16x16x4_f32(false, a0, false, bop,
                                                   (short)0, zero8, false, false);
    int   s1l = 16 + (lane & 15);
    float ux = __shfl(px, s1l, 32), uy = __shfl(py, s1l, 32), uz = __shfl(pz, s1l, 32);
    v2f a1; a1.x = (lane < 16) ? ux : uz; a1.y = (lane < 16) ? uy : 1.0f;
    v8f d1 = __builtin_amdgcn_wmma_f32_16x16x4_f32(false, a1, false, bop,
                                                   (short)0, zero8, false, false);

    // redistribute D through per-wave LDS tile (D[m][n]: m<8 -> vgpr m @ lane n,
    // m>=8 -> vgpr m-8 @ lane n+16)
    float* wp = &s_wm[wv][lane][0];
    #pragma unroll
    for (int r = 0; r < 8; ++r) { wp[r] = d0[r]; wp[8 + r] = d1[r]; }

    // wait async staging + make LDS visible block-wide
    asm volatile("s_wait_asynccnt 0" ::: "memory");
    __syncthreads();

    const int m    = lane & 15;
    const int ridx = (m & 7) + ((lane >= 16) ? 8 : 0);
    const int ls   = (m < 8) ? 0 : 16;
    float mx = s_wm[wv][ls + 0][ridx];
    float my = s_wm[wv][ls + 1][ridx];
    float mz = s_wm[wv][ls + 2][ridx];

    // ---- per-gaussian attributes from staged LDS ----
    const float* rg = &s_raw[tid * 18];
    float sr0 = rg[0], sr1 = rg[1];
    float q0 = rg[2], q1 = rg[3], q2 = rg[4], q3 = rg[5];

    // scales
    float sig0 = fast_rcp(1.0f + fast_exp(-sr0));
    float sig1 = fast_rcp(1.0f + fast_exp(-sr1));
    float smul = depth * mult;
    float sc0 = (0.5f + 14.5f * sig0) * smul;
    float sc1 = (0.5f + 14.5f * sig1) * smul;

    // rotations (normalized with eps)
    float qn  = fast_sqrt(q0 * q0 + q1 * q1 + q2 * q2 + q3 * q3) + 1e-8f;
    float iqn = fast_rcp(qn);
    float rw = q0 * iqn, rx = q1 * iqn, ry = q2 * iqn, rz = q3 * iqn;

    // world rotation quat = rot * c2w_q, then quat->rotmat (normalized)
    float gw = rw * qcw - rx * qcx - ry * qcy - rz * qcz;
    float gx = rw * qcx + rx * qcw + ry * qcz - rz * qcy;
    float gy = rw * qcy - rx * qcz + ry * qcw + rz * qcx;
    float gz = rw * qcz + rx * qcy - ry * qcx + rz * qcw;
    float gi = fast_rsq(gw * gw + gx * gx + gy * gy + gz * gz);
    gw *= gi; gx *= gi; gy *= gi; gz *= gi;
    float G00 = 1.0f - 2.0f * (gy * gy + gz * gz);
    float G01 = 2.0f * (gx * gy - gw * gz);
    float G02 = 2.0f * (gx * gz + gw * gy);
    float G10 = 2.0f * (gx * gy + gw * gz);
    float G11 = 1.0f - 2.0f * (gx * gx + gz * gz);
    float G12 = 2.0f * (gy * gz - gw * gx);
    float G20 = 2.0f * (gx * gz - gw * gy);
    float G21 = 2.0f * (gy * gz + gw * gx);
    float G22 = 1.0f - 2.0f * (gx * gx + gy * gy);

    // ---- stores ----
    const size_t NN = (size_t)N;
    // means (3N): stage in LDS, drain via async LDS->global b128 (coalesced)
    s_means[tid * 3 + 0] = mx;
    s_means[tid * 3 + 1] = my;
    s_means[tid * 3 + 2] = mz;
    // covariances (16N): rows 0..2 = RS|0 with RS[i][j]=Rg[j][i]*s3[i], row3 = means|1
    float4* covp = (float4*)(out + 3 * NN) + (size_t)gid * 4;
    covp[0] = make_float4(G00 * sc0, G10 * sc0, G20 * sc0, 0.0f);
    covp[1] = make_float4(G01 * sc1, G11 * sc1, G21 * sc1, 0.0f);
    covp[2] = make_float4(G02,       G12,       G22,       0.0f);
    covp[3] = make_float4(mx, my, mz, 1.0f);
    // scales (2N)
    ((float2*)(out + 19 * NN))[gid] = make_float2(sc0, sc1);
    // rotations (4N)
    ((float4*)(out + 21 * NN))[gid] = make_float4(rw, rx, ry, rz);
    // harmonics (12N): D1[i][j] = R[perm[i]][perm[j]], perm = [1,2,0]
    float4* hp = (float4*)(out + 25 * NN) + (size_t)gid * 3;
    #pragma unroll
    for (int ch = 0; ch < 3; ++ch) {
        float h0 = rg[6 + ch * 4 + 0];
        float b1 = rg[6 + ch * 4 + 1] * 0.025f;
        float b2 = rg[6 + ch * 4 + 2] * 0.025f;
        float b3 = rg[6 + ch * 4 + 3] * 0.025f;
        float o1 = R11 * b1 + R12 * b2 + R10 * b3;
        float o2 = R21 * b1 + R22 * b2 + R20 * b3;
        float o3 = R01 * b1 + R02 * b2 + R00 * b3;
        hp[ch] = make_float4(h0, o1, o2, o3);
    }
    // opacities (N)
    out[37 * NN + (size_t)gid] = opac[gid];

    // drain means slab: LDS -> global, async b128 (threads 0..191; per-wave
    // uniform trip counts so EXEC stays full). Completion is guaranteed by the
    // implicit S_WAIT_IDLE at S_ENDPGM.
    __syncthreads();
    float* gdst = out + (size_t)blockIdx.x * (256 * 3);
    #pragma unroll 1
    for (int i = tid; i < (256 * 3) / 4; i += 256) {
        uint32_t loff = (uint32_t)(uintptr_t)(&s_means[i * 4]);
        float*   ga   = gdst + (size_t)i * 4;
        asm volatile("global_store_async_from_lds_b128 %0, %1, off"
                     :: "v"(ga), "v"(loff) : "memory");
    }
}

// ---------------------------------------------------------------------------
extern "C" void kernel_launch(void* const* d_in, const int* in_sizes, int n_in,
                              void* d_out, int out_size, void* d_ws, size_t ws_size,
                              hipStream_t stream)
{
    const float* extr   = (const float*)d_in[0];
    const float* intr   = (const float*)d_in[1];
    const float* coords = (const float*)d_in[2];
    const float* depths = (const float*)d_in[3];
    const float* opacs  = (const float*)d_in[4];
    const float* raw    = (const float*)d_in[5];
    const int*   Hp     = (const int*)d_in[6];
    const int*   Wp     = (const int*)d_in[7];

    const int N       = in_sizes[3];          // B*V*R*SRF*SPP = 1048576
    const int nviews  = in_sizes[0] / 16;     // B*V = 16
    const int perView = N / nviews;           // 65536
    float* vp = (float*)d_ws;

    hipLaunchKernelGGL(gs_view_precompute, dim3(1), dim3(nviews), 0, stream,
                       extr, intr, Hp, Wp, vp, nviews);
    hipLaunchKernelGGL(gs_adapter_kernel, dim3(N / 256), dim3(256), 0, stream,
                       coords, depths, opacs, raw, vp, (float*)d_out, perView, N);
}